// GCN_6Layers_10574209483127
// MI455X (gfx1250) — compile-verified
//
#include <hip/hip_runtime.h>
#include <hip/hip_bf16.h>

#define N_NODES 10000
#define N_EDGES 640000
#define D 128

typedef __attribute__((ext_vector_type(2))) float v2f;
typedef __attribute__((ext_vector_type(8))) float v8f;

// ---------------------------------------------------------------------------
// Degree accumulation: out_deg[src] += 1, in_deg[dst] += 1
// ---------------------------------------------------------------------------
__global__ void gcn_degree_kernel(const int* __restrict__ src,
                                  const int* __restrict__ dst,
                                  float* __restrict__ out_deg,
                                  float* __restrict__ in_deg) {
    unsigned e = blockIdx.x * blockDim.x + threadIdx.x;
    if (e < N_EDGES) {
        atomicAdd(&out_deg[src[e]], 1.0f);
        atomicAdd(&in_deg[dst[e]], 1.0f);
    }
}

// deg -> rsqrt(max(deg,1))  (in place over 2*N_NODES floats)
__global__ void gcn_norm_kernel(float* __restrict__ deg) {
    unsigned i = blockIdx.x * blockDim.x + threadIdx.x;
    if (i < 2 * N_NODES) {
        deg[i] = rsqrtf(fmaxf(deg[i], 1.0f));
    }
}

// ---------------------------------------------------------------------------
// Edge scatter (SpMM): agg[dst] += in[src] * norm_src[src]
// One wave (32 lanes) per edge; 4 consecutive floats per lane (float4 gather).
// ---------------------------------------------------------------------------
__global__ void gcn_scatter_kernel(const float* __restrict__ in,
                                   const float* __restrict__ norm_src,
                                   const int* __restrict__ src,
                                   const int* __restrict__ dst,
                                   float* __restrict__ agg) {
    unsigned t = blockIdx.x * blockDim.x + threadIdx.x;
    unsigned e = t >> 5;            // edge index
    unsigned c = (t & 31u) * 4u;    // feature chunk start
    if (e >= N_EDGES) return;
    int s = src[e];
    int d = dst[e];
    float ns = norm_src[s];
    const float4 v = *reinterpret_cast<const float4*>(in + (size_t)s * D + c);
    float* ap = agg + (size_t)d * D + c;
    atomicAdd(ap + 0, v.x * ns);
    atomicAdd(ap + 1, v.y * ns);
    atomicAdd(ap + 2, v.z * ns);
    atomicAdd(ap + 3, v.w * ns);
}

// ---------------------------------------------------------------------------
// Fused GEMM: out = act( (agg * norm_dst[:,None]) @ W + b )
// Full-precision f32 matrix core: V_WMMA_F32_16X16X4_F32.
// One wave computes a 16-row x 128-col output strip: 8 accumulator tiles,
// A fragment reused across all 8 N-tiles per K-step.
// blockDim = (32,4): 4 waves per block, 625 strips total.
// ---------------------------------------------------------------------------
__global__ void gcn_gemm_kernel(const float* __restrict__ agg,
                                const float* __restrict__ norm_dst,
                                const float* __restrict__ W,
                                const float* __restrict__ bias,
                                float* __restrict__ out,
                                int do_relu) {
    const int tile = blockIdx.x * 4 + threadIdx.y;   // row-strip id, uniform per wave
    if (tile >= (N_NODES + 15) / 16) return;

    const int lane = threadIdx.x;        // 0..31
    const int half = lane >> 4;          // 0: K even pair low, 1: K pair high
    const int l16  = lane & 15;

    // A-side row for this lane (clamped to keep EXEC all-ones for WMMA)
    int row = tile * 16 + l16;
    int rowc = row < N_NODES ? row : (N_NODES - 1);
    const float ascale = norm_dst[rowc];
    const float* __restrict__ arow = agg + (size_t)rowc * D;

    v8f acc[8];
#pragma unroll
    for (int tn = 0; tn < 8; ++tn) acc[tn] = (v8f){0,0,0,0,0,0,0,0};

    for (int k0 = 0; k0 < D; k0 += 4) {
        const int ka = k0 + half * 2;    // this lane's K pair: ka, ka+1
        v2f a;
        a.x = arow[ka]     * ascale;
        a.y = arow[ka + 1] * ascale;
        const float* __restrict__ wr0 = W + (size_t)ka * D;        // row ka
        const float* __restrict__ wr1 = wr0 + D;                   // row ka+1
#pragma unroll
        for (int tn = 0; tn < 8; ++tn) {
            const int col = tn * 16 + l16;
            v2f b;
            b.x = wr0[col];
            b.y = wr1[col];
            acc[tn] = __builtin_amdgcn_wmma_f32_16x16x4_f32(
                false, a, false, b, (short)0, acc[tn], false, false);
        }
    }

    // Epilogue: bias + optional ReLU + store (bounds-checked)
#pragma unroll
    for (int tn = 0; tn < 8; ++tn) {
        const int col = tn * 16 + l16;
        const float bv = bias[col];
#pragma unroll
        for (int r = 0; r < 8; ++r) {
            int orow = tile * 16 + r + half * 8;
            if (orow < N_NODES) {
                float v = acc[tn][r] + bv;
                if (do_relu) v = fmaxf(v, 0.0f);
                out[(size_t)orow * D + col] = v;
            }
        }
    }
}

// ---------------------------------------------------------------------------
// Launch: degrees/norms once, then 6 x (memset agg -> scatter -> wmma gemm).
// Outputs land directly in d_out slices ordered (h6,h5,h4,h3,h2,h1).
// ---------------------------------------------------------------------------
extern "C" void kernel_launch(void* const* d_in, const int* in_sizes, int n_in,
                              void* d_out, int out_size, void* d_ws, size_t ws_size,
                              hipStream_t stream) {
    const float* x   = (const float*)d_in[0];
    const int*   src = (const int*)d_in[1];
    const int*   dst = (const int*)d_in[2];
    const float* Wm[6];
    const float* bm[6];
    for (int i = 0; i < 6; ++i) {
        Wm[i] = (const float*)d_in[3 + 2 * i];
        bm[i] = (const float*)d_in[4 + 2 * i];
    }
    float* out = (float*)d_out;

    // Workspace layout: agg [N*D] | norm_src [N] | norm_dst [N]
    float* agg      = (float*)d_ws;
    float* norm_src = agg + (size_t)N_NODES * D;
    float* norm_dst = norm_src + N_NODES;

    // Degrees -> norms (norm_src/norm_dst are contiguous: one memset, one finalize)
    hipMemsetAsync(norm_src, 0, 2 * N_NODES * sizeof(float), stream);
    gcn_degree_kernel<<<(N_EDGES + 255) / 256, 256, 0, stream>>>(src, dst, norm_src, norm_dst);
    gcn_norm_kernel<<<(2 * N_NODES + 255) / 256, 256, 0, stream>>>(norm_src);

    const unsigned scatter_blocks = (unsigned)(((size_t)N_EDGES * 32 + 255) / 256);
    const int strips = (N_NODES + 15) / 16;            // 625
    const int gemm_blocks = (strips + 3) / 4;          // 4 waves (tiles) per block

    const float* in = x;
    for (int l = 0; l < 6; ++l) {
        hipMemsetAsync(agg, 0, (size_t)N_NODES * D * sizeof(float), stream);
        gcn_scatter_kernel<<<scatter_blocks, 256, 0, stream>>>(in, norm_src, src, dst, agg);
        float* hout = out + (size_t)(5 - l) * N_NODES * D;   // h(l+1) slice
        gcn_gemm_kernel<<<gemm_blocks, dim3(32, 4), 0, stream>>>(
            agg, norm_dst, Wm[l], bm[l], hout, (l < 5) ? 1 : 0);
        in = hout;
    }
}